// WeightQuantizer_36206574305852
// MI455X (gfx1250) — compile-verified
//
#include <hip/hip_runtime.h>
#include <stdint.h>

// WeightQuantizer (NVFP4-style groupwise quant-dequant) for MI455X / gfx1250.
//
// Streaming kernel: 64 MiB in + 64 MiB out @ 23.3 TB/s  ->  ~5.8 us floor.
// No matmul => WMMA is inapplicable. CDNA5 paths used instead:
//   * global_load_async_to_lds_b128 double-buffered pipeline (ASYNCcnt)
//   * s_wait_asynccnt / s_wait_dscnt split counters
//   * non-temporal b128 stores (write-once output)
//   * wave32 layout: one 16-float group per lane, amax fully intra-lane.
// v3: 80B-padded LDS slots (stride 20 dwords, gcd(20,64)=4 -> all 64 banks hit
//     evenly; the old 64B stride hit only 16 banks = 4x ds_load_b128 slowdown);
//     (em+0x100)&0x7E00 carry-round + v_min_u32 clamp; max3-shaped amax tree.

typedef float v4f __attribute__((ext_vector_type(4)));

// Nearest E4M3 (normal-only grid [2^-7, 480]), ties -> lower value
// (matches jnp.argmin-first over the enumerated grid).
static __device__ __forceinline__ float e4m3_round_pos(float s) {
    uint32_t b = __float_as_uint(s);
    b += 0x7FFFFu;            // round f32 mantissa to 3 bits; exact ties do NOT carry -> round down
    b &= 0xFFF00000u;
    float r = __uint_as_float(b);
    return fminf(fmaxf(r, 0.0078125f), 480.0f);
}

// quant_fp4 bit-twiddle (fp16 -> E2M1 value set, round-half-away, clamp 6) + dequant.
static __device__ __forceinline__ float fp4_qdq(float x, float kk, float scale) {
    const float GM = 1.0f / 6.0f;                  // float(GLOBAL_MAX)
    _Float16 h = (_Float16)(x * kk);               // ~ (x/GM)/scale, then fp16 RNE
    uint32_t bi = (uint32_t)__builtin_bit_cast(uint16_t, h);
    uint32_t sg = bi & 0x8000u;
    uint32_t em = bi & 0x7FFFu;
    // >=1.0 region: mantissa-carry round to E2M1 ((em&0x7E00)+2*(em&0x100) ==
    // (em+0x100)&0x7E00), then clamp to 6.0 (also covers inf/NaN like the ref).
    uint32_t hi = (em + 0x100u) & 0x7E00u;
    hi = hi < 0x4600u ? hi : 0x4600u;              // v_min_u32
    // <1.0 region: {0, 0.5, 1.0} with thresholds 0.25 / 0.75.
    uint32_t lo = em >= 0x3A00u ? 0x3C00u : (em >= 0x3400u ? 0x3800u : 0u);
    uint32_t r  = em < 0x3C00u ? lo : hi;
    float qf = (float)__builtin_bit_cast(_Float16, (uint16_t)(r | sg));
    return (scale * qf) * GM;                      // scale*qf is exact; one final rounding (matches ref)
}

static __device__ __forceinline__ void process_group(
    const v4f a0, const v4f a1, const v4f a2, const v4f a3,
    v4f& o0, v4f& o1, v4f& o2, v4f& o3)
{
    const float GM = 1.0f / 6.0f;
    // max3-shaped tree -> v_max3_num_f32 with folded |x| modifiers.
    float m0 = fmaxf(fmaxf(fabsf(a0.x), fabsf(a0.y)), fabsf(a0.z));
    float m1 = fmaxf(fmaxf(fabsf(a0.w), fabsf(a1.x)), fabsf(a1.y));
    float m2 = fmaxf(fmaxf(fabsf(a1.z), fabsf(a1.w)), fabsf(a2.x));
    float m3 = fmaxf(fmaxf(fabsf(a2.y), fabsf(a2.z)), fabsf(a2.w));
    float m4 = fmaxf(fmaxf(fabsf(a3.x), fabsf(a3.y)), fabsf(a3.z));
    float t0 = fmaxf(fmaxf(m0, m1), m2);
    float t1 = fmaxf(fmaxf(m3, m4), fabsf(a3.w));
    float m  = fmaxf(t0, t1);
    float scale = e4m3_round_pos((m / GM) / 6.0f); // == amax, e4m3-rounded
    float kk = 6.0f / scale;                       // amortize both per-elem divisions
    o0.x = fp4_qdq(a0.x, kk, scale); o0.y = fp4_qdq(a0.y, kk, scale);
    o0.z = fp4_qdq(a0.z, kk, scale); o0.w = fp4_qdq(a0.w, kk, scale);
    o1.x = fp4_qdq(a1.x, kk, scale); o1.y = fp4_qdq(a1.y, kk, scale);
    o1.z = fp4_qdq(a1.z, kk, scale); o1.w = fp4_qdq(a1.w, kk, scale);
    o2.x = fp4_qdq(a2.x, kk, scale); o2.y = fp4_qdq(a2.y, kk, scale);
    o2.z = fp4_qdq(a2.z, kk, scale); o2.w = fp4_qdq(a2.w, kk, scale);
    o3.x = fp4_qdq(a3.x, kk, scale); o3.y = fp4_qdq(a3.y, kk, scale);
    o3.z = fp4_qdq(a3.z, kk, scale); o3.w = fp4_qdq(a3.w, kk, scale);
}

// Issue one lane's 64B group as 4 async B128 copies global->LDS (ASYNCcnt += 4).
// IOFFSET is added to BOTH the LDS and global addresses (ISA 08, 4.4), so one
// (vdst_lds, vaddr, saddr) triple covers the whole group — and stays correct
// with the 80B-padded LDS slots, since only the per-lane bases differ.
static __device__ __forceinline__ void async_load_group(uint32_t ldsaddr, uint32_t goff,
                                                        unsigned long long base) {
    asm volatile(
        "s_wait_dscnt 0x0\n\t"   // prior ds_loads from this buffer fully drained (WAR on LDS)
        "global_load_async_to_lds_b128 %0, %1, %2 offset:0\n\t"
        "global_load_async_to_lds_b128 %0, %1, %2 offset:16\n\t"
        "global_load_async_to_lds_b128 %0, %1, %2 offset:32\n\t"
        "global_load_async_to_lds_b128 %0, %1, %2 offset:48"
        :: "v"(ldsaddr), "v"(goff), "s"(base)
        : "memory");
}

__global__ __launch_bounds__(256, 1) void WeightQuantizer_36206574305852_kernel(
    const float* __restrict__ x, float* __restrict__ out,
    long long G, unsigned int stride, int Kfull)
{
    // 2 buffers x 256 lanes x (64B data + 16B pad) = 40 KB of 320 KB/WGP LDS.
    __shared__ v4f smem[2][256][5];

    const uint32_t tid = blockIdx.x * blockDim.x + threadIdx.x;

    const unsigned long long base = (unsigned long long)(uintptr_t)x;
    // Low 32 bits of a generic shared pointer == wave-relative LDS byte address.
    uint32_t ldsA[2];
    ldsA[0] = (uint32_t)(uintptr_t)&smem[0][threadIdx.x][0];
    ldsA[1] = (uint32_t)(uintptr_t)&smem[1][threadIdx.x][0];

    if (Kfull > 0) {
        async_load_group(ldsA[0], tid * 64u, base);              // prologue: tile 0
        for (int k = 0; k < Kfull; ++k) {
            if (k + 1 < Kfull) {                                 // uniform branch
                uint32_t goff = (uint32_t)(((unsigned long long)tid +
                                 (unsigned long long)(k + 1) * stride) * 64ull);
                async_load_group(ldsA[(k + 1) & 1], goff, base); // prefetch tile k+1
                asm volatile("s_wait_asynccnt 0x4" ::: "memory"); // tile k landed in LDS
            } else {
                asm volatile("s_wait_asynccnt 0x0" ::: "memory");
            }
            const v4f* sp = &smem[k & 1][threadIdx.x][0];
            v4f a0 = sp[0], a1 = sp[1], a2 = sp[2], a3 = sp[3];  // ds_load_b128 x4, conflict-free
            v4f o0, o1, o2, o3;
            process_group(a0, a1, a2, a3, o0, o1, o2, o3);
            long long g = (long long)tid + (long long)k * stride;
            v4f* op = (v4f*)(out + g * 16);
            __builtin_nontemporal_store(o0, op + 0);             // th:TH_STORE_NT b128
            __builtin_nontemporal_store(o1, op + 1);
            __builtin_nontemporal_store(o2, op + 2);
            __builtin_nontemporal_store(o3, op + 3);
        }
    }

    // Tail (empty for 4096x4096 with the launch below): direct-load path.
    for (long long g = (long long)tid + (long long)Kfull * stride; g < G;
         g += (long long)stride) {
        const v4f* ip = (const v4f*)(x + g * 16);
        v4f a0 = ip[0], a1 = ip[1], a2 = ip[2], a3 = ip[3];
        v4f o0, o1, o2, o3;
        process_group(a0, a1, a2, a3, o0, o1, o2, o3);
        v4f* op = (v4f*)(out + g * 16);
        __builtin_nontemporal_store(o0, op + 0);
        __builtin_nontemporal_store(o1, op + 1);
        __builtin_nontemporal_store(o2, op + 2);
        __builtin_nontemporal_store(o3, op + 3);
    }
}

extern "C" void kernel_launch(void* const* d_in, const int* in_sizes, int n_in,
                              void* d_out, int out_size, void* d_ws, size_t ws_size,
                              hipStream_t stream) {
    (void)n_in; (void)out_size; (void)d_ws; (void)ws_size;
    const float* x = (const float*)d_in[0];
    float* out = (float*)d_out;
    long long n = (long long)in_sizes[0];
    long long G = n / 16;                          // groups of 16 along last dim
    const int threads = 256;                       // 8 wave32s
    long long want = (G + threads - 1) / threads;
    int blocks = (int)(want < 1024 ? want : 1024); // 4096^2 -> 1024 blocks, Kfull = 4
    if (blocks < 1) blocks = 1;
    unsigned int stride = (unsigned int)blocks * (unsigned int)threads;
    int Kfull = (int)(G / (long long)stride);      // host-side division
    WeightQuantizer_36206574305852_kernel<<<blocks, threads, 0, stream>>>(
        x, out, G, stride, Kfull);
}